// GraphEncoder_7842610283501
// MI455X (gfx1250) — compile-verified
//
#include <hip/hip_runtime.h>

typedef float v2f __attribute__((ext_vector_type(2)));
typedef float v8f __attribute__((ext_vector_type(8)));

#define D 128  // feature dim for in/hid/out

// ---------------------------------------------------------------- utilities
__global__ __launch_bounds__(256) void fill_kernel(float* __restrict__ p, float v, long long n) {
    long long i = (long long)blockIdx.x * blockDim.x + threadIdx.x;
    long long stride = (long long)gridDim.x * blockDim.x;
    for (; i < n; i += stride) p[i] = v;
}

// deg[dst]++ per edge (deg pre-filled with 1.0 for the self loop)
__global__ __launch_bounds__(256) void deg_kernel(const int* __restrict__ dst, float* __restrict__ deg, int nE) {
    int i = blockIdx.x * blockDim.x + threadIdx.x;
    int stride = gridDim.x * blockDim.x;
    for (; i < nE; i += stride) atomicAdd(&deg[dst[i]], 1.0f);
}

__global__ __launch_bounds__(256) void rsqrt_kernel(float* __restrict__ deg, int n) {
    int i = blockIdx.x * blockDim.x + threadIdx.x;
    if (i < n) deg[i] = rsqrtf(deg[i]);  // deg >= 1 always (self loop)
}

// ---------------------------------------------------------------- GEMM: C[n,128] = A[n,128] @ W[128,128]
// 256 threads = 8 waves; wave w computes rows [blk*128 + w*16, +16) x all 128 cols
// via V_WMMA_F32_16X16X4_F32.
//
// W is staged in LDS in a K-pair-interleaved, XOR-32 swizzled layout:
//   element (k, n) lives at float index  (k>>1)*256 + ((2n + (k&1)) ^ (((k>>1)&1)<<5))
// so that a B fragment {W[kk][n], W[kk+1][n]} (kk even) is ONE aligned ds_load_b64
// straight into a consecutive VGPR pair (no repack movs before v_wmma), and since
// kp&1 == (lane>>4) on every K step, the ^32 swizzle maps lanes 0-15 onto banks
// 0-31 and lanes 16-31 onto banks 32-63: conflict-free with zero padding.
__global__ __launch_bounds__(256) void gemm_wmma_f32(const float* __restrict__ A,
                                                     const float* __restrict__ W,
                                                     float* __restrict__ C, int n) {
    __shared__ float wlds[D * D];  // 64 KB
    // cooperative load + relayout of W (row-major [K][N] in global)
    for (int i = threadIdx.x; i < D * D / 4; i += 256) {
        int k  = i >> 5;          // 32 float4 per row
        int n0 = (i & 31) << 2;   // starting col of this float4
        float4 w4 = ((const float4*)W)[i];
        int base = (k >> 1) * 256 + (k & 1);
        int sw   = ((k >> 1) & 1) << 5;
        wlds[base + (((n0 + 0) << 1) ^ sw)] = w4.x;
        wlds[base + (((n0 + 1) << 1) ^ sw)] = w4.y;
        wlds[base + (((n0 + 2) << 1) ^ sw)] = w4.z;
        wlds[base + (((n0 + 3) << 1) ^ sw)] = w4.w;
    }
    __syncthreads();

    const int wave = threadIdx.x >> 5;
    const int lane = threadIdx.x & 31;
    const int m    = lane & 15;
    const int half = lane >> 4;
    const int row0 = blockIdx.x * 128 + wave * 16;

    int ar = row0 + m;
    if (ar >= n) ar = n - 1;  // clamp so WMMA runs with EXEC all-ones
    const float* arow = A + (size_t)ar * D;

    v8f acc[8];
#pragma unroll
    for (int t = 0; t < 8; ++t) acc[t] = (v8f){0.f, 0.f, 0.f, 0.f, 0.f, 0.f, 0.f, 0.f};

    const int sw = half << 5;  // kp&1 == half for every k0 (k0 is a multiple of 4)
    for (int k0 = 0; k0 < D; k0 += 4) {
        const int kk = k0 + 2 * half;  // A lanes 0-15 hold K={k0,k0+1}, lanes 16-31 K={k0+2,k0+3}
        float2 av = *(const float2*)(arow + kk);
        v2f a;
        a.x = av.x;
        a.y = av.y;
        const float* bbase = wlds + (kk >> 1) * 256;
#pragma unroll
        for (int nt = 0; nt < 8; ++nt) {
            float2 bv = *(const float2*)(bbase + ((((nt * 16 + m) << 1)) ^ sw));
            v2f b;
            b.x = bv.x;  // W[kk][nt*16+m]
            b.y = bv.y;  // W[kk+1][nt*16+m]
            acc[nt] = __builtin_amdgcn_wmma_f32_16x16x4_f32(
                false, a, false, b, (short)0, acc[nt], false, false);
        }
    }

    // C/D layout: vgpr v, lanes 0-15 -> M=v, lanes 16-31 -> M=v+8; N = lane&15 within tile
#pragma unroll
    for (int nt = 0; nt < 8; ++nt) {
#pragma unroll
        for (int v = 0; v < 8; ++v) {
            int r = row0 + v + 8 * half;
            if (r < n) C[(size_t)r * D + nt * 16 + m] = acc[nt][v];
        }
    }
}

// ---------------------------------------------------------------- edge scatter-add
// one wave per edge: out[dst] += dinv[src]*dinv[dst] * xw[src]
__global__ __launch_bounds__(256) void scatter_edges(const int* __restrict__ src,
                                                     const int* __restrict__ dst,
                                                     const float* __restrict__ dinv,
                                                     const float* __restrict__ xw,
                                                     float* __restrict__ out, int nE) {
    int wave  = (blockIdx.x * blockDim.x + threadIdx.x) >> 5;
    int lane  = threadIdx.x & 31;
    int waves = (gridDim.x * blockDim.x) >> 5;
    for (int e = wave; e < nE; e += waves) {
        int s = src[e];
        int d = dst[e];
        float norm = dinv[s] * dinv[d];
        float4 v = ((const float4*)(xw + (size_t)s * D))[lane];
        float* o = out + (size_t)d * D + lane * 4;
        atomicAdd(o + 0, norm * v.x);
        atomicAdd(o + 1, norm * v.y);
        atomicAdd(o + 2, norm * v.z);
        atomicAdd(o + 3, norm * v.w);
    }
}

// out[i,f] = out[i,f] + dinv[i]^2 * xw[i,f] + b[f]  (optional ReLU) — self-loop + bias epilogue
__global__ __launch_bounds__(256) void finish_nodes(float* __restrict__ out,
                                                    const float* __restrict__ xw,
                                                    const float* __restrict__ dinv,
                                                    const float* __restrict__ b,
                                                    long long total, int relu) {
    long long t = (long long)blockIdx.x * blockDim.x + threadIdx.x;
    if (t >= total) return;
    int i = (int)(t >> 7);
    int f = (int)(t & 127);
    float di = dinv[i];
    float v = out[t] + di * di * xw[t] + b[f];
    if (relu) v = fmaxf(v, 0.0f);
    out[t] = v;
}

// ---------------------------------------------------------------- pooling
__global__ __launch_bounds__(256) void pool_scatter(const float* __restrict__ h,
                                                    const int* __restrict__ batch,
                                                    float* __restrict__ pool,
                                                    float* __restrict__ cnt, int n) {
    int wave  = (blockIdx.x * blockDim.x + threadIdx.x) >> 5;
    int lane  = threadIdx.x & 31;
    int waves = (gridDim.x * blockDim.x) >> 5;
    for (int i = wave; i < n; i += waves) {
        int g = batch[i];
        float4 v = ((const float4*)(h + (size_t)i * D))[lane];
        float* o = pool + (size_t)g * D + lane * 4;
        atomicAdd(o + 0, v.x);
        atomicAdd(o + 1, v.y);
        atomicAdd(o + 2, v.z);
        atomicAdd(o + 3, v.w);
        if (lane == 0) atomicAdd(&cnt[g], 1.0f);
    }
}

__global__ __launch_bounds__(256) void pool_div(const float* __restrict__ pool,
                                                const float* __restrict__ cnt,
                                                float* __restrict__ out, long long total) {
    long long t = (long long)blockIdx.x * blockDim.x + threadIdx.x;
    if (t >= total) return;
    out[t] = pool[t] / fmaxf(cnt[t >> 7], 1.0f);
}

// ---------------------------------------------------------------- driver
extern "C" void kernel_launch(void* const* d_in, const int* in_sizes, int n_in,
                              void* d_out, int out_size, void* d_ws, size_t ws_size,
                              hipStream_t stream) {
    const float* x    = (const float*)d_in[0];
    const int*  edges = (const int*)d_in[1];
    const int*  batch = (const int*)d_in[2];
    const float* W1   = (const float*)d_in[3];
    const float* b1   = (const float*)d_in[4];
    const float* W2   = (const float*)d_in[5];
    const float* b2   = (const float*)d_in[6];
    float* out = (float*)d_out;

    const int N = in_sizes[0] / D;   // 100000
    const int E = in_sizes[1] / 2;   // 1600000
    const int G = out_size / D;      // 1024
    const int* src = edges;
    const int* dst = edges + E;

    // workspace layout (floats)
    float* ws   = (float*)d_ws;
    size_t nd   = (size_t)N * D;
    float* B0   = ws;                       // xw buffer (layer1 then layer2)
    float* B1   = B0 + nd;                  // aggregate/h buffer
    float* dinv = B1 + nd;                  // [N] deg -> dinv
    float* pool = dinv + N;                 // [G*D]
    float* cnt  = pool + (size_t)G * D;     // [G]

    const long long ndl = (long long)nd;
    const int gemm_blocks = (N + 127) / 128;

    // ---- degree + norm (shared by both layers)
    fill_kernel<<<1024, 256, 0, stream>>>(dinv, 1.0f, N);
    deg_kernel<<<2048, 256, 0, stream>>>(dst, dinv, E);
    rsqrt_kernel<<<(N + 255) / 256, 256, 0, stream>>>(dinv, N);

    // ---- layer 1: xw1 = x @ W1 ; aggregate ; +self-loop +b1 ; ReLU
    gemm_wmma_f32<<<gemm_blocks, 256, 0, stream>>>(x, W1, B0, N);
    fill_kernel<<<4096, 256, 0, stream>>>(B1, 0.0f, ndl);
    scatter_edges<<<4096, 256, 0, stream>>>(src, dst, dinv, B0, B1, E);
    finish_nodes<<<(int)((ndl + 255) / 256), 256, 0, stream>>>(B1, B0, dinv, b1, ndl, 1);

    // ---- layer 2: xw2 = h1 @ W2 ; aggregate ; +self-loop +b2
    gemm_wmma_f32<<<gemm_blocks, 256, 0, stream>>>(B1, W2, B0, N);
    fill_kernel<<<4096, 256, 0, stream>>>(B1, 0.0f, ndl);
    scatter_edges<<<4096, 256, 0, stream>>>(src, dst, dinv, B0, B1, E);
    finish_nodes<<<(int)((ndl + 255) / 256), 256, 0, stream>>>(B1, B0, dinv, b2, ndl, 0);

    // ---- global mean pool
    const long long gl = (long long)G * D;
    fill_kernel<<<256, 256, 0, stream>>>(pool, 0.0f, gl);
    fill_kernel<<<8, 256, 0, stream>>>(cnt, 0.0f, G);
    pool_scatter<<<1024, 256, 0, stream>>>(B1, batch, pool, cnt, N);
    pool_div<<<(int)((gl + 255) / 256), 256, 0, stream>>>(pool, cnt, out, gl);
}